// GraphConvLayer_36275293782355
// MI455X (gfx1250) — compile-verified
//
#include <hip/hip_runtime.h>
#include <math.h>

#define NNODES 100000
#define NEDGES 1600000
#define D      128
#define EPSF   1e-5f

typedef __attribute__((ext_vector_type(2))) float v2f;
typedef __attribute__((ext_vector_type(8))) float v8f;

// ---------------------------------------------------------------- zero fill
__global__ void gcn_zero_kernel(float* __restrict__ p, int n) {
    int i = blockIdx.x * blockDim.x + threadIdx.x;
    int stride = gridDim.x * blockDim.x;
    for (; i < n; i += stride) p[i] = 0.0f;
}

// ------------------------------------------------------- degree scatter-add
__global__ void gcn_degree_kernel(const int* __restrict__ row,
                                  const float* __restrict__ adj,
                                  float* __restrict__ degree) {
    int e = blockIdx.x * blockDim.x + threadIdx.x;
    if (e < NEDGES) {
        unsafeAtomicAdd(&degree[row[e]], adj[e]);
    }
}

// --------------------------------------------------------------- d^-1/2
__global__ void gcn_dinv_kernel(const float* __restrict__ degree,
                                float* __restrict__ dinv) {
    int n = blockIdx.x * blockDim.x + threadIdx.x;
    if (n < NNODES) {
        float d = degree[n];
        d = (d == 0.0f) ? EPSF : d;
        dinv[n] = rsqrtf(d + EPSF);
    }
}

// ------------------------------------------------ SpMM scatter (wave/edge)
// One wave32 per edge; lane handles 4 contiguous channels (32*4 = 128).
// x[col] gather is a coalesced 512B read (hot in 192MB L2); scatter uses
// hardware global_atomic_add_f32.
__global__ __launch_bounds__(256)
void gcn_spmm_kernel(const float* __restrict__ x,
                     const int* __restrict__ row,
                     const int* __restrict__ col,
                     const float* __restrict__ adj,
                     const float* __restrict__ dinv,
                     float* __restrict__ agg) {
    long long tid = (long long)blockIdx.x * blockDim.x + threadIdx.x;
    int e    = (int)(tid >> 5);
    int lane = (int)(tid & 31);
    if (e >= NEDGES) return;

    int r = row[e];
    int c = col[e];
    float w = dinv[r] * adj[e] * dinv[c];

    const float4* xv = (const float4*)(x + (size_t)c * D);
    float4 v = xv[lane];

    float* dst = agg + (size_t)r * D + lane * 4;
    unsafeAtomicAdd(dst + 0, w * v.x);
    unsafeAtomicAdd(dst + 1, w * v.y);
    unsafeAtomicAdd(dst + 2, w * v.z);
    unsafeAtomicAdd(dst + 3, w * v.w);
}

// --------------------------------------------- GEMM: agg @ W + bias (WMMA)
// One wave32 per 16x16 output tile. K=128 -> 32 x v_wmma_f32_16x16x4_f32.
// Weight (64KB) staged in LDS once per block (8 waves cover all 8 N-tiles).
__global__ __launch_bounds__(256)
void gcn_gemm_kernel(const float* __restrict__ agg,
                     const float* __restrict__ weight,
                     const float* __restrict__ bias,
                     float* __restrict__ out) {
    __shared__ float wlds[D * D];   // 64 KB (WGP has 320 KB LDS)

    // cooperative float4 load of the whole 128x128 weight matrix
    for (int i = threadIdx.x; i < (D * D) / 4; i += 256) {
        ((float4*)wlds)[i] = ((const float4*)weight)[i];
    }
    __syncthreads();

    const int wave = threadIdx.x >> 5;           // 0..7
    const int lane = threadIdx.x & 31;           // wave32 lane
    const int tile = blockIdx.x * 8 + wave;      // 6250 blocks * 8 = 50000 tiles
    const int m0   = (tile >> 3) * 16;           // 6250 M-tiles (exact)
    const int n0   = (tile & 7) * 16;            // 8 N-tiles  (exact)

    const int half = lane >> 4;                  // 0: K even pair base, 1: +2
    const int l16  = lane & 15;                  // row (A) / col (B,D)

    v8f acc = {};

    const float* arow = agg + (size_t)(m0 + l16) * D;

    #pragma unroll
    for (int k0 = 0; k0 < D; k0 += 4) {
        const int ka = k0 + 2 * half;            // this lane's K pair base
        // A fragment: 16x4 tile, lane holds agg[m0+l16][ka], agg[m0+l16][ka+1]
        v2f a;
        a.x = arow[ka + 0];
        a.y = arow[ka + 1];
        // B fragment: 4x16 tile, lane holds W[ka][n0+l16], W[ka+1][n0+l16]
        v2f b;
        b.x = wlds[(ka + 0) * D + n0 + l16];
        b.y = wlds[(ka + 1) * D + n0 + l16];
        // D = A*B + C   (fp32 WMMA, 8 args: neg_a, A, neg_b, B, c_mod, C, reuse_a, reuse_b)
        acc = __builtin_amdgcn_wmma_f32_16x16x4_f32(false, a, false, b,
                                                    (short)0, acc, false, false);
    }

    const float bv = bias[n0 + l16];
    #pragma unroll
    for (int r = 0; r < 8; ++r) {
        // C/D layout: VGPR r -> M = r + 8*half, N = l16
        out[(size_t)(m0 + r + 8 * half) * D + n0 + l16] = acc[r] + bv;
    }
}

// --------------------------------------------------------------- launcher
extern "C" void kernel_launch(void* const* d_in, const int* in_sizes, int n_in,
                              void* d_out, int out_size, void* d_ws, size_t ws_size,
                              hipStream_t stream) {
    const float* x    = (const float*)d_in[0];
    const int*   row  = (const int*)d_in[1];
    const int*   col  = (const int*)d_in[2];
    const float* adj  = (const float*)d_in[3];
    const float* wgt  = (const float*)d_in[4];
    const float* bias = (const float*)d_in[5];
    float* out = (float*)d_out;

    // workspace layout: agg[NNODES*D] | degree[NNODES] | dinv[NNODES]
    float* agg    = (float*)d_ws;
    float* degree = agg + (size_t)NNODES * D;
    float* dinv   = degree + NNODES;

    // zero agg + degree every call (graph-replay deterministic)
    const int nzero = NNODES * D + NNODES;
    gcn_zero_kernel<<<4096, 256, 0, stream>>>(agg, nzero);

    gcn_degree_kernel<<<(NEDGES + 255) / 256, 256, 0, stream>>>(row, adj, degree);

    gcn_dinv_kernel<<<(NNODES + 255) / 256, 256, 0, stream>>>(degree, dinv);

    // one wave (32 threads) per edge -> 8 edges per 256-thread block
    gcn_spmm_kernel<<<NEDGES / 8, 256, 0, stream>>>(x, row, col, adj, dinv, agg);

    // 50000 tiles / 8 waves per block = 6250 blocks
    gcn_gemm_kernel<<<6250, 256, 0, stream>>>(agg, wgt, bias, out);
}